// ST_Transformer_super_gai_515396075934
// MI455X (gfx1250) — compile-verified
//
#include <hip/hip_runtime.h>
#include <hip/hip_bf16.h>

// ---------------- problem constants (match reference) ----------------
constexpr int N_SPOTS = 10000;
constexpr int D_LAT   = 128;
constexpr int D_IN    = 3000;
constexpr int KNN_K   = 10;
constexpr int SCALE   = 6;
constexpr float BN_EPS = 1e-4f;

// output layout (floats): z_lr | de_feat | x1_hr
constexpr long long Z_OFF  = 0;
constexpr long long DE_OFF = (long long)N_SPOTS * D_LAT;                   // 1,280,000
constexpr long long HR_OFF = DE_OFF + (long long)N_SPOTS * D_IN;           // 31,280,000

// workspace layout (bytes)
constexpr size_t WS_SUMS_OFF  = 0;            // float[2*D_IN]  (sum, sumsq)
constexpr size_t WS_STATS_OFF = 24000;        // float[2*D_IN]  (mean, rstd)
constexpr size_t WS_IDX_OFF   = 48000;        // int[N_SPOTS*KNN_K]

typedef __attribute__((ext_vector_type(2))) float v2f;
typedef __attribute__((ext_vector_type(8))) float v8f;

// ---------------------------------------------------------------------
// Kernel 1: brute-force KNN on 2D spatial coords, top-10 smallest d2.
// One thread per query; candidates tiled through LDS. Top-K kept in
// registers via fully-unrolled insertion (constant indices only).
// ---------------------------------------------------------------------
__global__ void knn_kernel(const float* __restrict__ spatial,
                           int* __restrict__ idx_out, int n) {
  __shared__ float sx[256];
  __shared__ float sy[256];
  const int q = blockIdx.x * blockDim.x + threadIdx.x;
  float qx = 0.f, qy = 0.f;
  if (q < n) { qx = spatial[2 * q]; qy = spatial[2 * q + 1]; }

  float bd[KNN_K];
  int   bi[KNN_K];
#pragma unroll
  for (int s = 0; s < KNN_K; ++s) { bd[s] = 3.0e38f; bi[s] = 0; }

  for (int t0 = 0; t0 < n; t0 += 256) {
    const int j = t0 + threadIdx.x;
    if (j < n) { sx[threadIdx.x] = spatial[2 * j]; sy[threadIdx.x] = spatial[2 * j + 1]; }
    __syncthreads();
    const int lim = min(256, n - t0);
    for (int jj = 0; jj < lim; ++jj) {
      const int jg = t0 + jj;
      const float dx = qx - sx[jj];
      const float dy = qy - sy[jj];
      const float d2 = dx * dx + dy * dy;
      if (jg != q && d2 < bd[KNN_K - 1]) {
        bd[KNN_K - 1] = d2;
        bi[KNN_K - 1] = jg;
        // one bubble pass restores sorted order (single new element)
#pragma unroll
        for (int s = KNN_K - 1; s > 0; --s) {
          if (bd[s] < bd[s - 1]) {
            float td = bd[s]; bd[s] = bd[s - 1]; bd[s - 1] = td;
            int   ti = bi[s]; bi[s] = bi[s - 1]; bi[s - 1] = ti;
          }
        }
      }
    }
    __syncthreads();
  }
  if (q < n) {
#pragma unroll
    for (int s = 0; s < KNN_K; ++s) idx_out[q * KNN_K + s] = bi[s];
  }
}

// ---------------------------------------------------------------------
// Kernel 2: softmax(-mean_d (x[idx]-x)^2) weights, then weighted
// gather aggregation for z_lr (D_LAT) and x1_hr (SCALE*D_LAT).
// One workgroup (128 threads) per row; thread d owns latent dim d.
// ---------------------------------------------------------------------
__global__ void agg_kernel(const float* __restrict__ x,
                           const float* __restrict__ xnbr,
                           const int* __restrict__ idxk,
                           float* __restrict__ z_lr,
                           float* __restrict__ x1_hr, int n) {
  const int row = blockIdx.x;
  const int d   = threadIdx.x;             // 0..127
  __shared__ float part[KNN_K][D_LAT];
  __shared__ float sums[KNN_K];
  __shared__ float wsm[KNN_K];

  int   idx[KNN_K];
  float xv[KNN_K];
  const float xd = x[row * D_LAT + d];
#pragma unroll
  for (int k = 0; k < KNN_K; ++k) {
    idx[k] = idxk[row * KNN_K + k];
    xv[k]  = x[idx[k] * D_LAT + d];
    const float df = xv[k] - xd;
    part[k][d] = df * df;
  }
  __syncthreads();

  if (d < KNN_K) {
    float s = 0.f;
    for (int j = 0; j < D_LAT; ++j) s += part[d][j];
    sums[d] = s * (1.0f / D_LAT);          // mean over latent dim
  }
  __syncthreads();

  if (d == 0) {
    float mx = -3.0e38f;
#pragma unroll
    for (int k = 0; k < KNN_K; ++k) mx = fmaxf(mx, -sums[k]);
    float e[KNN_K];
    float tot = 0.f;
#pragma unroll
    for (int k = 0; k < KNN_K; ++k) { e[k] = __expf(-sums[k] - mx); tot += e[k]; }
    const float inv = 1.0f / tot;
#pragma unroll
    for (int k = 0; k < KNN_K; ++k) wsm[k] = e[k] * inv;
  }
  __syncthreads();

  float w[KNN_K];
#pragma unroll
  for (int k = 0; k < KNN_K; ++k) w[k] = wsm[k];

  // LR aggregation (reuses register-cached gathers from the distance pass)
  float acc = 0.f;
#pragma unroll
  for (int k = 0; k < KNN_K; ++k) acc += w[k] * xv[k];
  z_lr[row * D_LAT + d] = acc;

  // HR aggregation: xn[n, s*D+d] = x_neighbor[s*N + n, d]
#pragma unroll
  for (int s = 0; s < SCALE; ++s) {
    float a = 0.f;
#pragma unroll
    for (int k = 0; k < KNN_K; ++k)
      a += w[k] * xnbr[((long long)(s * n + idx[k])) * D_LAT + d];
    x1_hr[(long long)row * (SCALE * D_LAT) + s * D_LAT + d] = a;
  }
}

// ---------------------------------------------------------------------
// Kernel 3: fp32 WMMA GEMM  H = z_lr[10000x128] @ W[128x3000] + bias.
// Each wave owns MBLK=5 M-tiles x one 16-wide N-tile and shares one B
// fragment across 5 independent V_WMMA_F32_16X16X4_F32 per k-step
// (different accumulators -> XDL pipelining, 5x less B traffic).
// M = 10000 = 125 * (5*16): no M guards. N-edge handled by clamping the
// load column (always in-bounds) and masking only the final stores, so
// the hot loop has no exec-mask manipulation.
// ---------------------------------------------------------------------
constexpr int MBLK = 5;

__global__ void gemm_wmma_kernel(const float* __restrict__ A,   // z_lr
                                 const float* __restrict__ W,   // [128,3000]
                                 const float* __restrict__ bias,
                                 float* __restrict__ H, int M) {
  const int lane = threadIdx.x & 31;
  const int wave = threadIdx.x >> 5;
  const int ntn  = (D_IN + 15) / 16;          // 188 N-tiles
  const int mgrp = M / (16 * MBLK);           // 125 M groups
  const int total = mgrp * ntn;               // 23,500 wave-tiles
  const int t = blockIdx.x * 8 + wave;
  if (t >= total) return;                     // wave-uniform exit

  const int mg = t / ntn;
  const int nt = t % ntn;
  const int m0 = mg * (16 * MBLK);
  const int n0 = nt * 16;

  const int colr = n0 + (lane & 15);          // real column (may be >= D_IN)
  const int col  = min(colr, D_IN - 1);       // clamped: loads always legal
  const int kb   = (lane >> 4) << 1;          // 0 or 2: K-pair within chunk
  const int rA   = m0 + (lane & 15);

  const float* __restrict__ Ap = A + (long long)rA * D_LAT + kb;
  const float* __restrict__ Wp = W + (long long)kb * D_IN + col;

  v8f c[MBLK] = {};

#pragma unroll 4
  for (int k0 = 0; k0 < D_LAT; k0 += 4) {
    v2f b;
    b.x = Wp[(long long)k0 * D_IN];
    b.y = Wp[(long long)(k0 + 1) * D_IN];
#pragma unroll
    for (int i = 0; i < MBLK; ++i) {
      v2f a;
      a.x = Ap[i * 16 * D_LAT + k0];
      a.y = Ap[i * 16 * D_LAT + k0 + 1];
      // 8 args: (neg_a, A, neg_b, B, c_mod, C, reuse_a, reuse_b)
      c[i] = __builtin_amdgcn_wmma_f32_16x16x4_f32(false, a, false, b,
                                                   (short)0, c[i], false, false);
    }
  }

  const float bv = bias[col];
  const int rbase = m0 + ((lane >> 4) << 3);
  if (colr < D_IN) {
#pragma unroll
    for (int i = 0; i < MBLK; ++i) {
#pragma unroll
      for (int r = 0; r < 8; ++r)
        H[(long long)(rbase + i * 16 + r) * D_IN + colr] = c[i][r] + bv;
    }
  }
}

// ---------------------------------------------------------------------
// Kernels 4-7: batch-norm statistics (train-mode batch stats) + ELU.
// ---------------------------------------------------------------------
__global__ void zero_ws_kernel(float* __restrict__ p, int n) {
  const int i = blockIdx.x * blockDim.x + threadIdx.x;
  if (i < n) p[i] = 0.f;
}

__global__ void col_partial_kernel(const float* __restrict__ H,
                                   float* __restrict__ sums, int M) {
  const int c = blockIdx.x * blockDim.x + threadIdx.x;
  if (c >= D_IN) return;
  const int chunk = (M + 31) / 32;
  const int r0 = blockIdx.y * chunk;
  const int r1 = min(M, r0 + chunk);
  float s = 0.f, s2 = 0.f;
  for (int r = r0; r < r1; ++r) {
    const float h = H[(long long)r * D_IN + c];
    s += h;
    s2 += h * h;
  }
  atomicAdd(&sums[c], s);
  atomicAdd(&sums[D_IN + c], s2);
}

__global__ void col_finalize_kernel(const float* __restrict__ sums,
                                    float* __restrict__ stats, int M) {
  const int c = blockIdx.x * blockDim.x + threadIdx.x;
  if (c >= D_IN) return;
  const float invM = 1.0f / (float)M;
  const float mu  = sums[c] * invM;
  const float var = sums[D_IN + c] * invM - mu * mu;   // biased (jnp.var)
  stats[c]        = mu;
  stats[D_IN + c] = rsqrtf(var + BN_EPS);
}

__global__ void bn_elu_kernel(float* __restrict__ H,
                              const float* __restrict__ stats,
                              const float* __restrict__ gamma,
                              const float* __restrict__ beta, long long total) {
  const long long i = (long long)blockIdx.x * blockDim.x + threadIdx.x;
  if (i >= total) return;
  const int c = (int)(i % D_IN);
  const float h = (H[i] - stats[c]) * stats[D_IN + c] * gamma[c] + beta[c];
  H[i] = h > 0.f ? h : expm1f(h);
}

// ---------------------------------------------------------------------
extern "C" void kernel_launch(void* const* d_in, const int* in_sizes, int n_in,
                              void* d_out, int out_size, void* d_ws, size_t ws_size,
                              hipStream_t stream) {
  const float* x      = (const float*)d_in[0];
  const float* xnbr   = (const float*)d_in[1];
  const float* spatial= (const float*)d_in[2];
  const float* W_dec  = (const float*)d_in[3];
  const float* b_dec  = (const float*)d_in[4];
  const float* gamma  = (const float*)d_in[5];
  const float* beta   = (const float*)d_in[6];

  float* out   = (float*)d_out;
  float* z_lr  = out + Z_OFF;
  float* de    = out + DE_OFF;
  float* hr    = out + HR_OFF;

  char*  ws       = (char*)d_ws;
  float* ws_sums  = (float*)(ws + WS_SUMS_OFF);
  float* ws_stats = (float*)(ws + WS_STATS_OFF);
  int*   ws_idx   = (int*)(ws + WS_IDX_OFF);

  const int N = N_SPOTS;

  // 1) KNN graph (spatial)
  knn_kernel<<<(N + 255) / 256, 256, 0, stream>>>(spatial, ws_idx, N);

  // 2) softmax weights + gather aggregation -> z_lr, x1_hr
  agg_kernel<<<N, D_LAT, 0, stream>>>(x, xnbr, ws_idx, z_lr, hr, N);

  // 3) decoder Linear via fp32 WMMA -> raw h into de_feat region
  const int ntn = (D_IN + 15) / 16;                         // 188
  const int waves_total = (N / (16 * MBLK)) * ntn;          // 23,500
  gemm_wmma_kernel<<<(waves_total + 7) / 8, 256, 0, stream>>>(z_lr, W_dec, b_dec, de, N);

  // 4-6) batch statistics over rows
  zero_ws_kernel<<<(2 * D_IN + 255) / 256, 256, 0, stream>>>(ws_sums, 2 * D_IN);
  dim3 gpart((D_IN + 255) / 256, 32);
  col_partial_kernel<<<gpart, 256, 0, stream>>>(de, ws_sums, N);
  col_finalize_kernel<<<(D_IN + 255) / 256, 256, 0, stream>>>(ws_sums, ws_stats, N);

  // 7) normalize + ELU in place
  const long long total = (long long)N * D_IN;
  bn_elu_kernel<<<(int)((total + 255) / 256), 256, 0, stream>>>(de, ws_stats, gamma, beta, total);
}